// Actor_49108656062911
// MI455X (gfx1250) — compile-verified
//
#include <hip/hip_runtime.h>
#include <hip/hip_bf16.h>
#include <math.h>

// ---------------------------------------------------------------------------
// Actor forward for MI455X (gfx1250, wave32, WMMA).
//
// Design notes (compile-only reasoning):
//  * Total ~20 GFLOPs, ~16MB real input traffic -> must NOT materialize the
//    [B,NH,A,A] logits (256MB) nor [B,A,A,6] edge feats (192MB). Attention is
//    fully fused per (batch, 16-row tile, head).
//  * DH == 4 maps exactly onto V_WMMA_F32_16X16X4_F32 (16x4 @ 4x16), fp32
//    exact.  All 32-wide GEMMs (QKV/O/MLP) chain the same instruction.
//  * Attention hot loop: per-batch positions staged in LDS (broadcast reads),
//    K/V kept as pipelined per-wave global loads (L2-resident) with
//    global_prefetch_b8 ahead of the stream.
// ---------------------------------------------------------------------------

typedef __attribute__((ext_vector_type(2))) float v2f;
typedef __attribute__((ext_vector_type(8))) float v8f;

#define BDIM 8
#define ADIM 1024
#define HID 32
#define NH 8
#define DH 4
#define EPS 1e-5f

__device__ __forceinline__ v8f wmma4(v2f a, v2f b, v8f c) {
  // D = A(16x4) * B(4x16) + C, fp32.
  // (neg_a, A, neg_b, B, c_mod, C, reuse_a, reuse_b)
  return __builtin_amdgcn_wmma_f32_16x16x4_f32(false, a, false, b, (short)0, c,
                                               false, false);
}

__device__ __forceinline__ float gelu_tanh(float x) {
  const float c = 0.7978845608028654f; // sqrt(2/pi)
  float t = tanhf(c * (x + 0.044715f * x * x * x));
  return 0.5f * x * (1.0f + t);
}

// ---------------------------------------------------------------------------
// Kernel 1: conv backbone.  One wave per sample; lane = output channel.
// BN applied as a per-channel LDS pass before each conv.  11->9->7->5->3->1.
// ---------------------------------------------------------------------------
struct ConvParams {
  const float* g[5];
  const float* bt[5];
  const float* m[5];
  const float* v[5];
  const float* w[5];
  const float* cb[5];
};

template <int HIN, int HOUT>
__device__ void conv32_layer(const float* __restrict__ wgt, float cbias,
                             float sc, float sh, float* src, float* dst,
                             int lane, bool relu) {
  const int HIN2 = HIN * HIN;
  // BN on this layer's input (lane owns channel == lane)
  for (int p = 0; p < HIN2; ++p)
    src[lane * HIN2 + p] = src[lane * HIN2 + p] * sc + sh;
  __syncthreads();

  float acc[HOUT * HOUT];
#pragma unroll
  for (int p = 0; p < HOUT * HOUT; ++p) acc[p] = cbias;

  for (int ic = 0; ic < 32; ++ic) {
    float wr[9];
#pragma unroll
    for (int t = 0; t < 9; ++t) wr[t] = wgt[lane * 288 + ic * 9 + t];
    const float* s = src + ic * HIN2;
#pragma unroll
    for (int y = 0; y < HOUT; ++y)
#pragma unroll
      for (int x = 0; x < HOUT; ++x) {
        float a = 0.f;
#pragma unroll
        for (int ky = 0; ky < 3; ++ky)
#pragma unroll
          for (int kx = 0; kx < 3; ++kx)
            a += s[(y + ky) * HIN + (x + kx)] * wr[ky * 3 + kx];
        acc[y * HOUT + x] += a;
      }
  }
#pragma unroll
  for (int p = 0; p < HOUT * HOUT; ++p) {
    float o = acc[p];
    if (relu) o = fmaxf(o, 0.f);
    dst[lane * HOUT * HOUT + p] = o;
  }
  __syncthreads();
}

__global__ __launch_bounds__(64) void conv_kernel(
    const float* __restrict__ obs, ConvParams P, float* __restrict__ hout) {
  __shared__ float buf[2][2][2592];
  const int wid = threadIdx.x >> 5;
  const int lane = threadIdx.x & 31;
  const int s = blockIdx.x * 2 + wid;
  float* A = buf[wid][0];
  float* Bb = buf[wid][1];

  // ---- layer 0: BN0 applied at load (4 input channels), conv 4->32, 11->9
  float sc0[4], sh0[4];
#pragma unroll
  for (int c = 0; c < 4; ++c) {
    float iv = rsqrtf(P.v[0][c] + EPS);
    sc0[c] = P.g[0][c] * iv;
    sh0[c] = P.bt[0][c] - P.m[0][c] * sc0[c];
  }
  for (int t = lane; t < 484; t += 32) {
    int c = t / 121;
    A[t] = obs[(size_t)s * 484 + t] * sc0[c] + sh0[c];
  }
  __syncthreads();

  float wr0[36];
#pragma unroll
  for (int t = 0; t < 36; ++t) wr0[t] = P.w[0][lane * 36 + t];
  float cb0 = P.cb[0][lane];
#pragma unroll 1
  for (int y = 0; y < 9; ++y)
    for (int x = 0; x < 9; ++x) {
      float a = cb0;
#pragma unroll
      for (int ic = 0; ic < 4; ++ic)
#pragma unroll
        for (int ky = 0; ky < 3; ++ky)
#pragma unroll
          for (int kx = 0; kx < 3; ++kx)
            a += A[ic * 121 + (y + ky) * 11 + (x + kx)] *
                 wr0[ic * 9 + ky * 3 + kx];
      Bb[lane * 81 + y * 9 + x] = fmaxf(a, 0.f);
    }
  __syncthreads();

  // ---- layers 1..4 (32->32)
  {
    float sc = P.g[1][lane] * rsqrtf(P.v[1][lane] + EPS);
    float sh = P.bt[1][lane] - P.m[1][lane] * sc;
    conv32_layer<9, 7>(P.w[1], P.cb[1][lane], sc, sh, Bb, A, lane, true);
  }
  {
    float sc = P.g[2][lane] * rsqrtf(P.v[2][lane] + EPS);
    float sh = P.bt[2][lane] - P.m[2][lane] * sc;
    conv32_layer<7, 5>(P.w[2], P.cb[2][lane], sc, sh, A, Bb, lane, true);
  }
  {
    float sc = P.g[3][lane] * rsqrtf(P.v[3][lane] + EPS);
    float sh = P.bt[3][lane] - P.m[3][lane] * sc;
    conv32_layer<5, 3>(P.w[3], P.cb[3][lane], sc, sh, Bb, A, lane, true);
  }
  {
    float sc = P.g[4][lane] * rsqrtf(P.v[4][lane] + EPS);
    float sh = P.bt[4][lane] - P.m[4][lane] * sc;
    conv32_layer<3, 1>(P.w[4], P.cb[4][lane], sc, sh, A, Bb, lane, false);
  }
  hout[(size_t)s * 32 + lane] = Bb[lane];
}

// ---------------------------------------------------------------------------
// Kernel 2: LN1 + QKV projections via chained V_WMMA_F32_16X16X4_F32.
// One wave per 16-row tile.  48 wmma / wave.
// ---------------------------------------------------------------------------
__global__ __launch_bounds__(32) void qkv_kernel(
    const float* __restrict__ h, const float* __restrict__ lg,
    const float* __restrict__ lb, const float* __restrict__ wq,
    const float* __restrict__ bq, const float* __restrict__ wk,
    const float* __restrict__ bk, const float* __restrict__ wv,
    const float* __restrict__ bv, float* __restrict__ Q, float* __restrict__ K,
    float* __restrict__ V) {
  __shared__ float hb[16 * 32];
  __shared__ float muS[16], rsS[16], glS[32], blS[32];
  const int lane = threadIdx.x;
  const int base = blockIdx.x * 16;

  for (int t = lane; t < 512; t += 32) hb[t] = h[(size_t)base * 32 + t];
  glS[lane] = lg[lane];
  blS[lane] = lb[lane];
  __syncthreads();

  if (lane < 16) {
    float sm = 0.f;
#pragma unroll
    for (int c = 0; c < 32; ++c) sm += hb[lane * 32 + c];
    float mu = sm * (1.f / 32.f);
    float vv = 0.f;
#pragma unroll
    for (int c = 0; c < 32; ++c) {
      float d = hb[lane * 32 + c] - mu;
      vv += d * d;
    }
    muS[lane] = mu;
    rsS[lane] = rsqrtf(vv * (1.f / 32.f) + EPS);
  }
  __syncthreads();

  const int l15 = lane & 15;
  const int half = lane >> 4;

  v2f a[8];
#pragma unroll
  for (int kb = 0; kb < 8; ++kb) {
    int k0 = kb * 4 + 2 * half;
    float mu = muS[l15], rs = rsS[l15];
    a[kb][0] = (hb[l15 * 32 + k0] - mu) * rs * glS[k0] + blS[k0];
    a[kb][1] = (hb[l15 * 32 + k0 + 1] - mu) * rs * glS[k0 + 1] + blS[k0 + 1];
  }

  const float* Ws[3] = {wq, wk, wv};
  const float* Bs[3] = {bq, bk, bv};
  float* Os[3] = {Q, K, V};
  for (int mtx = 0; mtx < 3; ++mtx) {
    for (int nt2 = 0; nt2 < 2; ++nt2) {
      int nt = nt2 * 16;
      v8f acc = {0.f, 0.f, 0.f, 0.f, 0.f, 0.f, 0.f, 0.f};
#pragma unroll
      for (int kb = 0; kb < 8; ++kb) {
        int k0 = kb * 4 + 2 * half;
        v2f bf;
        bf[0] = Ws[mtx][k0 * 32 + nt + l15];
        bf[1] = Ws[mtx][(k0 + 1) * 32 + nt + l15];
        acc = wmma4(a[kb], bf, acc);
      }
      float bc = Bs[mtx][nt + l15];
#pragma unroll
      for (int j = 0; j < 8; ++j)
        Os[mtx][(size_t)(base + j + 8 * half) * 32 + nt + l15] = acc[j] + bc;
    }
  }
}

// ---------------------------------------------------------------------------
// Kernel 3: fused attention (+edge bias/mask/softmax/PV) + O-proj + LN2 + MLP.
// Grid: 512 workgroups = (batch, 16-row i-tile). 8 waves = 8 heads.
// Logits via WMMA f32 16x16x4 (K = DH = 4 exactly).  PV kept lane-distributed
// and reduced once at the end via __shfl_xor over the 16-lane column group.
// exp() without max-subtraction: masked entries contribute exactly 0 and
// unmasked logits are tiny (weights ~0.1), denominator > 0 (self always in).
// Batch positions staged once in LDS; K/V prefetched with global_prefetch_b8.
// ---------------------------------------------------------------------------
__global__ __launch_bounds__(256) void attn_block_kernel(
    const float* __restrict__ Qp, const float* __restrict__ Kp,
    const float* __restrict__ Vp, const int* __restrict__ pos,
    const float* __restrict__ ew, const float* __restrict__ ebias,
    const float* __restrict__ wo, const float* __restrict__ bo,
    const float* __restrict__ l2g, const float* __restrict__ l2b,
    const float* __restrict__ w1, const float* __restrict__ b1,
    const float* __restrict__ w2, const float* __restrict__ b2,
    float* __restrict__ h) {
  __shared__ float posl[2 * ADIM];  // this batch's positions, as float
  __shared__ float ctxs[16 * 32];
  __shared__ float xs[16 * 32];
  __shared__ float m1s[16 * 128];
  __shared__ float mu2[16], rs2[16], g2l[32], b2l[32];

  const int wid = threadIdx.x >> 5;  // wave id == head
  const int lane = threadIdx.x & 31;
  const int l15 = lane & 15;
  const int half = lane >> 4;
  const int b = blockIdx.x >> 6;
  const int it = blockIdx.x & 63;
  const int rowbase = b * ADIM + it * 16;  // global row (0..8191)

  // stage positions of this batch into LDS (broadcast reads in the hot loop)
  for (int t = threadIdx.x; t < 2 * ADIM; t += 256)
    posl[t] = (float)pos[(size_t)b * 2 * ADIM + t];
  if (threadIdx.x < 32) {
    g2l[threadIdx.x] = l2g[threadIdx.x];
    b2l[threadIdx.x] = l2b[threadIdx.x];
  }
  __syncthreads();

  const int head = wid;
  // Q A-frag (scaled by 1/sqrt(DH) = 0.5)
  v2f aq;
  {
    const float2 qq =
        *(const float2*)&Qp[(size_t)(rowbase + l15) * 32 + 4 * head + 2 * half];
    aq[0] = qq.x * 0.5f;
    aq[1] = qq.y * 0.5f;
  }
  // row positions for this lane's 8 D-rows
  float pix[8], piy[8];
#pragma unroll
  for (int j = 0; j < 8; ++j) {
    int r = it * 16 + j + 8 * half;  // row within batch
    pix[j] = posl[r * 2 + 0];
    piy[j] = posl[r * 2 + 1];
  }
  // edge weights for this head (reference produces 5 features: dx,dy,linf,l1,l2)
  const float we0 = ew[0 * NH + head], we1 = ew[1 * NH + head];
  const float we2 = ew[2 * NH + head], we3 = ew[3 * NH + head];
  const float we4 = ew[4 * NH + head];
  const float eb = ebias[head];
  const float inv_l1 = 1.0f / 10.0f;         // h+w = 10
  const float inv_l2 = 0.1414213562373095f;  // 1/sqrt(50)

  float lsum[8], c0[8], c1[8], c2[8], c3[8];
#pragma unroll
  for (int j = 0; j < 8; ++j) {
    lsum[j] = 0.f;
    c0[j] = c1[j] = c2[j] = c3[j] = 0.f;
  }

  for (int jt = 0; jt < 64; ++jt) {
    const int jrow = b * ADIM + jt * 16 + l15;
    // keep upcoming K/V tiles warm in the near cache (global_prefetch_b8)
    if (jt + 4 < 64) {
      const int prow = jrow + 64;
      __builtin_prefetch(&Kp[(size_t)prow * 32 + 4 * head], 0, 3);
      __builtin_prefetch(&Vp[(size_t)prow * 32 + 4 * head], 0, 3);
    }
    v2f bf;
    {
      const float2 kk =
          *(const float2*)&Kp[(size_t)jrow * 32 + 4 * head + 2 * half];
      bf[0] = kk.x;
      bf[1] = kk.y;
    }
    v8f zz = {0.f, 0.f, 0.f, 0.f, 0.f, 0.f, 0.f, 0.f};
    v8f s = wmma4(aq, bf, zz);
    const float4 vj = *(const float4*)&Vp[(size_t)jrow * 32 + 4 * head];
    const int jl = (jt * 16 + l15) * 2;  // within-batch position index
    const float pjx = posl[jl + 0];
    const float pjy = posl[jl + 1];
#pragma unroll
    for (int j = 0; j < 8; ++j) {
      float dx = pix[j] - pjx, dy = piy[j] - pjy;
      float adx = fabsf(dx), ady = fabsf(dy);
      float linf = fmaxf(adx, ady);
      float l1 = (adx + ady) * inv_l1;
      float l2 = sqrtf(dx * dx + dy * dy) * inv_l2;
      float bias = dx * we0 + dy * we1 + linf * we2 + l1 * we3 + l2 * we4 + eb;
      float p = (linf <= 7.0f) ? __expf(s[j] + bias) : 0.f;
      lsum[j] += p;
      c0[j] += p * vj.x;
      c1[j] += p * vj.y;
      c2[j] += p * vj.z;
      c3[j] += p * vj.w;
    }
  }

  // reduce over the 16-lane column group (xor masks 1..8 stay inside halves)
#pragma unroll
  for (int j = 0; j < 8; ++j) {
    for (int mk = 1; mk < 16; mk <<= 1) {
      lsum[j] += __shfl_xor(lsum[j], mk);
      c0[j] += __shfl_xor(c0[j], mk);
      c1[j] += __shfl_xor(c1[j], mk);
      c2[j] += __shfl_xor(c2[j], mk);
      c3[j] += __shfl_xor(c3[j], mk);
    }
  }
  if (l15 < 4) {
#pragma unroll
    for (int j = 0; j < 8; ++j) {
      float val = (l15 == 0) ? c0[j] : (l15 == 1) ? c1[j] : (l15 == 2) ? c2[j]
                                                                       : c3[j];
      ctxs[(j + 8 * half) * 32 + 4 * head + l15] = val / lsum[j];
    }
  }
  __syncthreads();

  // ---- O projection + residual -> xs (waves 0,1 cover N-tiles)
  if (wid < 2) {
    const int nt = wid * 16;
    v2f a[8];
#pragma unroll
    for (int kb = 0; kb < 8; ++kb) {
      int k0 = kb * 4 + 2 * half;
      a[kb][0] = ctxs[l15 * 32 + k0];
      a[kb][1] = ctxs[l15 * 32 + k0 + 1];
    }
    v8f acc = {0.f, 0.f, 0.f, 0.f, 0.f, 0.f, 0.f, 0.f};
#pragma unroll
    for (int kb = 0; kb < 8; ++kb) {
      int k0 = kb * 4 + 2 * half;
      v2f bf;
      bf[0] = wo[k0 * 32 + nt + l15];
      bf[1] = wo[(k0 + 1) * 32 + nt + l15];
      acc = wmma4(a[kb], bf, acc);
    }
    float bc = bo[nt + l15];
#pragma unroll
    for (int j = 0; j < 8; ++j) {
      int r = j + 8 * half;
      float x = h[(size_t)(rowbase + r) * 32 + nt + l15] + acc[j] + bc;
      xs[r * 32 + nt + l15] = x;
    }
  }
  __syncthreads();

  // ---- LN2 stats
  if (threadIdx.x < 16) {
    int r = threadIdx.x;
    float sm = 0.f;
#pragma unroll
    for (int c = 0; c < 32; ++c) sm += xs[r * 32 + c];
    float mu = sm * (1.f / 32.f);
    float vv = 0.f;
#pragma unroll
    for (int c = 0; c < 32; ++c) {
      float d = xs[r * 32 + c] - mu;
      vv += d * d;
    }
    mu2[r] = mu;
    rs2[r] = rsqrtf(vv * (1.f / 32.f) + EPS);
  }
  __syncthreads();

  // ---- MLP1 (32 -> 128) + gelu, each wave owns one N-tile of 16
  {
    const int nt = wid * 16;
    v2f a[8];
#pragma unroll
    for (int kb = 0; kb < 8; ++kb) {
      int k0 = kb * 4 + 2 * half;
      float mu = mu2[l15], rs = rs2[l15];
      a[kb][0] = (xs[l15 * 32 + k0] - mu) * rs * g2l[k0] + b2l[k0];
      a[kb][1] = (xs[l15 * 32 + k0 + 1] - mu) * rs * g2l[k0 + 1] + b2l[k0 + 1];
    }
    v8f acc = {0.f, 0.f, 0.f, 0.f, 0.f, 0.f, 0.f, 0.f};
#pragma unroll
    for (int kb = 0; kb < 8; ++kb) {
      int k0 = kb * 4 + 2 * half;
      v2f bf;
      bf[0] = w1[k0 * 128 + nt + l15];
      bf[1] = w1[(k0 + 1) * 128 + nt + l15];
      acc = wmma4(a[kb], bf, acc);
    }
    float bc = b1[nt + l15];
#pragma unroll
    for (int j = 0; j < 8; ++j)
      m1s[(j + 8 * half) * 128 + nt + l15] = gelu_tanh(acc[j] + bc);
  }
  __syncthreads();

  // ---- MLP2 (128 -> 32) + residual -> h (waves 0,1)
  if (wid < 2) {
    const int nt = wid * 16;
    v8f acc = {0.f, 0.f, 0.f, 0.f, 0.f, 0.f, 0.f, 0.f};
#pragma unroll
    for (int kb = 0; kb < 32; ++kb) {
      int k0 = kb * 4 + 2 * half;
      v2f a, bf;
      a[0] = m1s[l15 * 128 + k0];
      a[1] = m1s[l15 * 128 + k0 + 1];
      bf[0] = w2[k0 * 32 + nt + l15];
      bf[1] = w2[(k0 + 1) * 32 + nt + l15];
      acc = wmma4(a, bf, acc);
    }
    float bc = b2[nt + l15];
#pragma unroll
    for (int j = 0; j < 8; ++j) {
      int r = j + 8 * half;
      h[(size_t)(rowbase + r) * 32 + nt + l15] =
          xs[r * 32 + nt + l15] + acc[j] + bc;
    }
  }
}

// ---------------------------------------------------------------------------
// Kernel 4: output head + action mask.
// ---------------------------------------------------------------------------
__global__ __launch_bounds__(256) void out_kernel(
    const float* __restrict__ h, const float* __restrict__ wout,
    const float* __restrict__ bout, const float* __restrict__ amask,
    float* __restrict__ out) {
  int t = blockIdx.x * blockDim.x + threadIdx.x;
  if (t >= 8192 * 5) return;
  int row = t / 5, act = t % 5;
  float s = bout[act];
#pragma unroll
  for (int c = 0; c < 32; ++c) s += h[(size_t)row * 32 + c] * wout[c * 5 + act];
  out[t] = s - 1e10f * (1.0f - amask[t]);
}

// ---------------------------------------------------------------------------
// Launch.  d_in flattened in setup_inputs() insertion order:
//  0 observations, 1 curr_positions, 2 action_masks,
//  3..22  bns[i]{g,b,m,v} i=0..4
//  23..32 convs[i]{w,b}
//  33..68 blocks[k]{ln1_g,ln1_b,ln2_g,ln2_b,(q,k,v,o){w,b},edge{w,b},
//                   mlp1{w,b},mlp2{w,b}}  (18 leaves each)
//  69 out.w, 70 out.b
// ---------------------------------------------------------------------------
extern "C" void kernel_launch(void* const* d_in, const int* in_sizes, int n_in,
                              void* d_out, int out_size, void* d_ws,
                              size_t ws_size, hipStream_t stream) {
  const float* obs = (const float*)d_in[0];
  const int* pos = (const int*)d_in[1];
  const float* amask = (const float*)d_in[2];

  ConvParams P;
  for (int i = 0; i < 5; ++i) {
    P.g[i] = (const float*)d_in[3 + 4 * i + 0];
    P.bt[i] = (const float*)d_in[3 + 4 * i + 1];
    P.m[i] = (const float*)d_in[3 + 4 * i + 2];
    P.v[i] = (const float*)d_in[3 + 4 * i + 3];
    P.w[i] = (const float*)d_in[23 + 2 * i];
    P.cb[i] = (const float*)d_in[23 + 2 * i + 1];
  }
  const float* outw = (const float*)d_in[69];
  const float* outb = (const float*)d_in[70];

  float* h = (float*)d_ws;
  float* Q = h + 8192 * 32;
  float* K = Q + 8192 * 32;
  float* V = K + 8192 * 32;

  conv_kernel<<<4096, 64, 0, stream>>>(obs, P, h);

  for (int blk = 0; blk < 2; ++blk) {
    const int bb = 33 + 18 * blk;
    const float* ln1g = (const float*)d_in[bb + 0];
    const float* ln1b = (const float*)d_in[bb + 1];
    const float* ln2g = (const float*)d_in[bb + 2];
    const float* ln2b = (const float*)d_in[bb + 3];
    const float* wq = (const float*)d_in[bb + 4];
    const float* bq = (const float*)d_in[bb + 5];
    const float* wk = (const float*)d_in[bb + 6];
    const float* bk = (const float*)d_in[bb + 7];
    const float* wv = (const float*)d_in[bb + 8];
    const float* bv = (const float*)d_in[bb + 9];
    const float* wo = (const float*)d_in[bb + 10];
    const float* bo = (const float*)d_in[bb + 11];
    const float* ew = (const float*)d_in[bb + 12];
    const float* ebv = (const float*)d_in[bb + 13];
    const float* w1 = (const float*)d_in[bb + 14];
    const float* b1 = (const float*)d_in[bb + 15];
    const float* w2 = (const float*)d_in[bb + 16];
    const float* b2 = (const float*)d_in[bb + 17];

    qkv_kernel<<<512, 32, 0, stream>>>(h, ln1g, ln1b, wq, bq, wk, bk, wv, bv,
                                       Q, K, V);
    attn_block_kernel<<<512, 256, 0, stream>>>(Q, K, V, pos, ew, ebv, wo, bo,
                                               ln2g, ln2b, w1, b1, w2, b2, h);
  }

  out_kernel<<<(8192 * 5 + 255) / 256, 256, 0, stream>>>(h, outw, outb, amask,
                                                         (float*)d_out);
}